// OpticalConv2dNew_75411035783258
// MI455X (gfx1250) — compile-verified
//
#include <hip/hip_runtime.h>
#include <hip/hip_bf16.h>

typedef __attribute__((ext_vector_type(2))) float v2f;
typedef __attribute__((ext_vector_type(8))) float v8f;

#define HN   224
#define HW   (HN * HN)        // 50176
#define HW4  (HW / 4)         // 12544
#define BSZ  16
#define COUT 64
#define CIN  64
#define KTAP 49               // 7x7
#define KPAD 52               // padded to multiple of 4

// ---------------------------------------------------------------------------
// Kernel 1: channel sum  S[b,y,x] = sum_ic x[b,ic,y,x]
// Pure bandwidth: 205 MB read, 3.2 MB write, float4 vectorized.
// ---------------------------------------------------------------------------
__global__ __launch_bounds__(256) void sum_channels_kernel(
    const float* __restrict__ x, float* __restrict__ s) {
  int p = blockIdx.x * blockDim.x + threadIdx.x;       // float4 index
  if (p >= BSZ * HW4) return;
  int b = p / HW4;
  int r = p - b * HW4;
  const float4* xp = reinterpret_cast<const float4*>(x) + (size_t)b * CIN * HW4 + r;
  float4 acc;
  acc.x = 0.f; acc.y = 0.f; acc.z = 0.f; acc.w = 0.f;
#pragma unroll 8
  for (int ic = 0; ic < CIN; ++ic) {
    float4 v = xp[(size_t)ic * HW4];
    acc.x += v.x; acc.y += v.y; acc.z += v.z; acc.w += v.w;
  }
  reinterpret_cast<float4*>(s)[(size_t)b * HW4 + r] = acc;
}

// ---------------------------------------------------------------------------
// Kernel 2: 7x7 circular conv via V_WMMA_F32_16X16X4_F32
//   out[b,oc,y,x] = sum_{u,v} K[oc,u,v] * S[b,(y+4-u)%224,(x+4-v)%224]
// Block: 128 threads (4 waves), one batch, 16x16 pixel patch, all 64 oc.
//   wave w -> oc group [16w, 16w+16)
// GEMM per row: M=16 oc (A = weights), N=16 pixels (B = S), K=52 (13 wmma).
// ---------------------------------------------------------------------------
__global__ __launch_bounds__(128) void conv7x7_wmma_kernel(
    const float* __restrict__ S, const float* __restrict__ Kw,
    float* __restrict__ out) {
  __shared__ float Wlds[KPAD][COUT];   // [k][oc], taps >= 49 zeroed (13.3 KB)
  __shared__ float Slds[22][24];       // halo patch rows y0-2..y0+19, cols x0-2..x0+19

  const int tid = threadIdx.x;
  const int x0  = blockIdx.x * 16;
  const int y0  = blockIdx.y * 16;
  const int b   = blockIdx.z;

  // --- load weights (kernel[oc, ic=0, :, :]) into LDS, zero-padded to K=52 ---
  for (int i = tid; i < KPAD * COUT; i += 128) {
    int oc = i / KPAD;
    int kk = i - oc * KPAD;
    Wlds[kk][oc] = (kk < KTAP) ? Kw[oc * (CIN * KTAP) + kk] : 0.0f;
  }
  // --- load S halo with circular wrap: 22x22 ---
  for (int i = tid; i < 22 * 22; i += 128) {
    int r = i / 22;
    int c = i - r * 22;
    int gy = (y0 - 2 + r + HN) % HN;
    int gx = (x0 - 2 + c + HN) % HN;
    Slds[r][c] = S[(size_t)b * HW + gy * HN + gx];
  }
  __syncthreads();

  const int wave = tid >> 5;
  const int lane = tid & 31;
  const int half = lane >> 4;     // K sub-block: lanes 0-15 -> K={0,1}, 16-31 -> K={2,3}
  const int li   = lane & 15;     // M index for A, N index for B
  const int oc0  = wave * 16;
  const int oc   = oc0 + li;

  // --- preload A (weights) per k-step: invariant over the 16 output rows ---
  v2f a[13];
  int soff0[12], soff1[12];       // LDS byte offsets for B loads (t=0..11 all valid)
#pragma unroll
  for (int t = 0; t < 13; ++t) {
    int k0 = 4 * t + 2 * half;
    a[t].x = Wlds[k0][oc];
    a[t].y = Wlds[k0 + 1][oc];
    if (t < 12) {
      int u0 = k0 / 7, v0 = k0 - 7 * u0;
      int u1 = (k0 + 1) / 7, v1 = (k0 + 1) - 7 * u1;
      // output row y reads Slds[yl + (6-u)][li + (6-v)]
      soff0[t] = ((6 - u0) * 24 + (li + 6 - v0)) * 4;
      soff1[t] = ((6 - u1) * 24 + (li + 6 - v1)) * 4;
    }
  }

  const char* sbase = reinterpret_cast<const char*>(&Slds[0][0]);
  const int   xw    = x0 + li;    // this lane's output x

  for (int yl = 0; yl < 16; ++yl) {
    v8f c = {0.f, 0.f, 0.f, 0.f, 0.f, 0.f, 0.f, 0.f};
    const char* srow = sbase + yl * (24 * 4);
#pragma unroll
    for (int t = 0; t < 12; ++t) {
      v2f bb;
      bb.x = *reinterpret_cast<const float*>(srow + soff0[t]);
      bb.y = *reinterpret_cast<const float*>(srow + soff1[t]);
      c = __builtin_amdgcn_wmma_f32_16x16x4_f32(
          /*neg_a=*/false, a[t], /*neg_b=*/false, bb,
          /*c_mod=*/(short)0, c, /*reuse_a=*/false, /*reuse_b=*/false);
    }
    // t = 12: absolute taps {48,49} (half=0) / {50,51} (half=1); only k=48 real
    {
      v2f bb;
      bb.x = half ? 0.f : Slds[yl][li];   // k=48: u=6,v=6 -> row yl+0, col li+0
      bb.y = 0.f;
      c = __builtin_amdgcn_wmma_f32_16x16x4_f32(
          false, a[12], false, bb, (short)0, c, false, false);
    }
    // --- store: D lane l, vgpr r -> (oc = oc0 + r + 8*half, x = x0 + li) ---
    const int y = y0 + yl;
#pragma unroll
    for (int r = 0; r < 8; ++r) {
      int occ = oc0 + r + 8 * half;
      out[(((size_t)b * COUT + occ) * HN + y) * HN + xw] = c[r];
    }
  }
}

// ---------------------------------------------------------------------------
extern "C" void kernel_launch(void* const* d_in, const int* in_sizes, int n_in,
                              void* d_out, int out_size, void* d_ws, size_t ws_size,
                              hipStream_t stream) {
  const float* x  = (const float*)d_in[0];   // [16,64,224,224] f32
  const float* kw = (const float*)d_in[1];   // [64,64,7,7] f32 (only ic=0 used)
  float* out = (float*)d_out;                // [16,64,224,224] f32
  float* S   = (float*)d_ws;                 // [16,224,224] f32 = 3.2 MB scratch

  // Kernel 1: channel-sum reduction
  int n4 = BSZ * HW4;
  sum_channels_kernel<<<(n4 + 255) / 256, 256, 0, stream>>>(x, S);

  // Kernel 2: WMMA circular conv
  dim3 grid(HN / 16, HN / 16, BSZ);   // (14, 14, 16)
  conv7x7_wmma_kernel<<<grid, 128, 0, stream>>>(S, kw, out);

  (void)in_sizes; (void)n_in; (void)out_size; (void)ws_size;
}